// GNNLayer_31447750542159
// MI455X (gfx1250) — compile-verified
//
#include <hip/hip_runtime.h>

#define N_USERS_C 50000
#define N_NODES_C 100000
#define N_EDGES_C 3200000

typedef __attribute__((ext_vector_type(2))) float v2f;
typedef __attribute__((ext_vector_type(8))) float v8f;

// ---------------- utility kernels ----------------

__global__ void fill_f32(float* __restrict__ p, float v, int n) {
    int i = blockIdx.x * blockDim.x + threadIdx.x;
    if (i < n) p[i] = v;
}

__global__ void deg_count(const int* __restrict__ dst, int E, float* __restrict__ deg) {
    int e = blockIdx.x * blockDim.x + threadIdx.x;
    if (e < E) atomicAdd(&deg[dst[e]], 1.0f);
}

__global__ void rsqrt_inplace(float* __restrict__ p, int n) {
    int i = blockIdx.x * blockDim.x + threadIdx.x;
    if (i < n) p[i] = rsqrtf(p[i]);
}

// ---------------- GEMM: h = X @ W ----------------
// K == 1 special case (layer 0): pure outer product.
__global__ void gemm_k1(const float* __restrict__ x, const float* __restrict__ W,
                        float* __restrict__ h, int F) {
    int idx = blockIdx.x * blockDim.x + threadIdx.x;
    int n = N_NODES_C * F;
    if (idx < n) {
        int i = idx / F;
        int j = idx - i * F;
        h[idx] = x[i] * W[j];
    }
}

// One wave (32 lanes) computes a 16x16 f32 tile via V_WMMA_F32_16X16X4_F32.
// A 16x4 f32 fragment: lanes 0-15 -> K=k0,k0+1 ; lanes 16-31 -> K=k0+2,k0+3 (M = lane%16).
// B 4x16 fragment: same K split, N = lane%16.
// C/D 16x16 f32: VGPR i -> row M = i + 8*(lane>=16), col N = lane%16.
__global__ __launch_bounds__(32)
void gemm_wmma_f32(const float* __restrict__ X, const float* __restrict__ W,
                   float* __restrict__ H, int K, int F) {
    const int tm = blockIdx.x;      // 16-row tile of M
    const int tn = blockIdx.y;      // 16-col tile of F
    const int lane = threadIdx.x;   // wave32
    const int half = lane >> 4;     // 0 or 1
    const int r    = lane & 15;

    const int row = tm * 16 + r;    // A-matrix row for this lane
    const int col = tn * 16 + r;    // B-matrix col for this lane

    v8f c = {};
    for (int k0 = 0; k0 < K; k0 += 4) {
        const int ka = k0 + 2 * half;
        v2f a, b;
        a.x = X[(size_t)row * K + ka];
        a.y = X[(size_t)row * K + ka + 1];
        b.x = W[(size_t)ka * F + col];
        b.y = W[(size_t)(ka + 1) * F + col];
        c = __builtin_amdgcn_wmma_f32_16x16x4_f32(
                /*neg_a=*/false, a, /*neg_b=*/false, b,
                /*c_mod=*/(short)0, c, /*reuse_a=*/false, /*reuse_b=*/false);
    }

    const int baseRow = tm * 16 + half * 8;
    const int outCol  = tn * 16 + r;
#pragma unroll
    for (int i = 0; i < 8; ++i) {
        H[(size_t)(baseRow + i) * F + outCol] = c[i];
    }
}

// ---------------- edge scatter: agg[dst] += h[src] * dinv[src]*dinv[dst] ----------------
// One thread per (edge, 4-float chunk).
__global__ void aggregate_edges(const int* __restrict__ src, const int* __restrict__ dst,
                                const float* __restrict__ dinv, const float* __restrict__ h,
                                float* __restrict__ agg, int E, int F) {
    const int chunks = F >> 2;
    int idx = blockIdx.x * blockDim.x + threadIdx.x;
    int tot = E * chunks;
    if (idx >= tot) return;
    int e = idx / chunks;
    int c = idx - e * chunks;
    int s = src[e];
    int d = dst[e];
    float w = dinv[s] * dinv[d];
    const float4 hv = *(const float4*)(h + (size_t)s * F + 4 * c);
    float* o = agg + (size_t)d * F + 4 * c;
    atomicAdd(o + 0, hv.x * w);
    atomicAdd(o + 1, hv.y * w);
    atomicAdd(o + 2, hv.z * w);
    atomicAdd(o + 3, hv.w * w);
}

// ---------------- finish: out = agg + h*dinv^2 (self loop) + bias (+relu) ----------------
__global__ void finish_rows(const float* __restrict__ agg, const float* __restrict__ h,
                            const float* __restrict__ dinv, const float* __restrict__ b,
                            float* __restrict__ out, int rowStart, int rowEnd, int F, int relu) {
    int idx = blockIdx.x * blockDim.x + threadIdx.x;
    int nrows = rowEnd - rowStart;
    if (idx >= nrows * F) return;
    int i = rowStart + idx / F;
    int j = idx % F;
    float di = dinv[i];
    size_t o = (size_t)i * F + j;
    float v = agg[o] + h[o] * di * di + b[j];
    if (relu) v = fmaxf(v, 0.0f);
    out[o] = v;
}

// ---------------- host orchestration ----------------

extern "C" void kernel_launch(void* const* d_in, const int* in_sizes, int n_in,
                              void* d_out, int out_size, void* d_ws, size_t ws_size,
                              hipStream_t stream) {
    (void)in_sizes; (void)n_in; (void)out_size; (void)ws_size;

    const float* x0 = (const float*)d_in[0];      // [N_NODES, 1]
    const int* eE = (const int*)d_in[1];          // edge_index  [2, E] flat
    const int* eU = (const int*)d_in[2];          // edge_index_u
    const int* eV = (const int*)d_in[3];          // edge_index_v

    const float* P[18];
    for (int i = 0; i < 18; ++i) P[i] = (const float*)d_in[4 + i];
    // layer L: Wu=P[6L], bu=P[6L+1], Wv=P[6L+2], bv=P[6L+3], Wuv=P[6L+4], buv=P[6L+5]

    float* ws = (float*)d_ws;
    const size_t NB = (size_t)N_NODES_C * 64;     // floats per node-feature buffer
    float* bufX = ws;                             // current layer input / next output
    float* bufH = ws + NB;                        // GEMM result
    float* bufG = ws + 2 * NB;                    // edge aggregation accumulator
    float* bufC = ws + 3 * NB;                    // concatenated [x_u ; x_v]
    float* dinvE = ws + 4 * NB;
    float* dinvU = dinvE + N_NODES_C;
    float* dinvV = dinvU + N_NODES_C;

    const int TB = 256;
    const int gN   = (N_NODES_C + TB - 1) / TB;
    const int gE   = (N_EDGES_C + TB - 1) / TB;

    // --- degrees / normalization (shared by all 3 layers, self-loop => init 1.0) ---
    fill_f32<<<gN, TB, 0, stream>>>(dinvE, 1.0f, N_NODES_C);
    fill_f32<<<gN, TB, 0, stream>>>(dinvU, 1.0f, N_NODES_C);
    fill_f32<<<gN, TB, 0, stream>>>(dinvV, 1.0f, N_NODES_C);
    deg_count<<<gE, TB, 0, stream>>>(eE + N_EDGES_C, N_EDGES_C, dinvE);
    deg_count<<<gE, TB, 0, stream>>>(eU + N_EDGES_C, N_EDGES_C, dinvU);
    deg_count<<<gE, TB, 0, stream>>>(eV + N_EDGES_C, N_EDGES_C, dinvV);
    rsqrt_inplace<<<gN, TB, 0, stream>>>(dinvE, N_NODES_C);
    rsqrt_inplace<<<gN, TB, 0, stream>>>(dinvU, N_NODES_C);
    rsqrt_inplace<<<gN, TB, 0, stream>>>(dinvV, N_NODES_C);

    const int dimsIn[3]  = {1, 64, 32};
    const int dimsOut[3] = {64, 32, 64};

    const float* xin = x0;   // layer-0 input has width 1

    for (int L = 0; L < 3; ++L) {
        const int K = dimsIn[L];
        const int F = dimsOut[L];
        const float *Wu = P[6*L], *bu = P[6*L+1], *Wv = P[6*L+2],
                    *bv = P[6*L+3], *Wuv = P[6*L+4], *buv = P[6*L+5];

        const int nf      = N_NODES_C * F;
        const int gNF     = (nf + TB - 1) / TB;
        const int gEdgeF  = (N_EDGES_C * (F / 4) + TB - 1) / TB;
        const dim3 gGemm(N_NODES_C / 16, F / 16);

        // ---- u-branch conv: rows [0, N_USERS) of result go to bufC ----
        if (K == 1) gemm_k1<<<gNF, TB, 0, stream>>>(xin, Wu, bufH, F);
        else        gemm_wmma_f32<<<gGemm, 32, 0, stream>>>(xin, Wu, bufH, K, F);
        fill_f32<<<gNF, TB, 0, stream>>>(bufG, 0.0f, nf);
        aggregate_edges<<<gEdgeF, TB, 0, stream>>>(eU, eU + N_EDGES_C, dinvU, bufH, bufG, N_EDGES_C, F);
        finish_rows<<<(N_USERS_C * F + TB - 1) / TB, TB, 0, stream>>>(
            bufG, bufH, dinvU, bu, bufC, 0, N_USERS_C, F, 0);

        // ---- v-branch conv: rows [N_USERS, N_NODES) of result go to bufC ----
        if (K == 1) gemm_k1<<<gNF, TB, 0, stream>>>(xin, Wv, bufH, F);
        else        gemm_wmma_f32<<<gGemm, 32, 0, stream>>>(xin, Wv, bufH, K, F);
        fill_f32<<<gNF, TB, 0, stream>>>(bufG, 0.0f, nf);
        aggregate_edges<<<gEdgeF, TB, 0, stream>>>(eV, eV + N_EDGES_C, dinvV, bufH, bufG, N_EDGES_C, F);
        finish_rows<<<((N_NODES_C - N_USERS_C) * F + TB - 1) / TB, TB, 0, stream>>>(
            bufG, bufH, dinvV, bv, bufC, N_USERS_C, N_NODES_C, F, 0);

        // ---- uv conv on concatenated features (F x F weight) ----
        gemm_wmma_f32<<<gGemm, 32, 0, stream>>>(bufC, Wuv, bufH, F, F);
        fill_f32<<<gNF, TB, 0, stream>>>(bufG, 0.0f, nf);
        aggregate_edges<<<gEdgeF, TB, 0, stream>>>(eE, eE + N_EDGES_C, dinvE, bufH, bufG, N_EDGES_C, F);

        float* outp = (L == 2) ? (float*)d_out : bufX;
        finish_rows<<<gNF, TB, 0, stream>>>(
            bufG, bufH, dinvE, buv, outp, 0, N_NODES_C, F, (L != 2) ? 1 : 0);

        xin = bufX;
    }
}